// PseudoLabel_Loss_39058432590479
// MI455X (gfx1250) — compile-verified
//
#include <hip/hip_runtime.h>
#include <hip/hip_bf16.h>
#include <math.h>

#define NCLS   5
#define NBINS  2048
#define HW     (512 * 512)
#define ENT_LO (-0.01f)
#define ENT_HI (1.70f)
#define EPS    (1e-6f)

typedef __attribute__((ext_vector_type(2))) float v2f;
typedef __attribute__((ext_vector_type(8))) float v8f;

// Wave32 sum-reduction using V_WMMA_F32_16X16X4_F32.
// A (16x4 f32): VGPR0 lanes0-15 -> A[m][0], lanes16-31 -> A[m][2]; VGPR1 = 0.
// B = ones(4x16). D[m][n] = A[m][0] + A[m][2] = x[m] + x[m+16], replicated over n.
// D layout: VGPR v, lane n<16 => D[v][n]; lane>=16 => D[v+8][n-16].
// So lanes<16 hold s[0..7] in d[0..7], lanes>=16 hold s[8..15]; sum 8 VGPRs,
// then one xor-16 shuffle-add gives the full 32-lane sum in every lane.
__device__ __forceinline__ float wave_sum_wmma(float x) {
    v2f a; a[0] = x;    a[1] = 0.0f;
    v2f b; b[0] = 1.0f; b[1] = 1.0f;
    v8f c = {};
    v8f d = __builtin_amdgcn_wmma_f32_16x16x4_f32(
        /*neg_a=*/false, a, /*neg_b=*/false, b,
        /*c_mod=*/(short)0, c, /*reuse_a=*/false, /*reuse_b=*/false);
    float h = d[0] + d[1] + d[2] + d[3] + d[4] + d[5] + d[6] + d[7];
    h += __shfl_xor(h, 16, 32);
    return h;
}

__global__ __launch_bounds__(256)
void k_zero(int* __restrict__ p, int n) {
    int i = blockIdx.x * blockDim.x + threadIdx.x;
    if (i < n) p[i] = 0;
}

// Pass A: entropy per pixel, per-class LDS histogram + per-class entropy sum.
template <bool STORE_ENT>
__global__ __launch_bounds__(256)
void k_ent_hist(const float* __restrict__ prob, const int* __restrict__ tgt,
                float* __restrict__ ent_out, int* __restrict__ ghist,
                float* __restrict__ gsum, int N) {
    __shared__ int lhist[NCLS * NBINS];   // 40 KB
    for (int i = threadIdx.x; i < NCLS * NBINS; i += blockDim.x) lhist[i] = 0;
    __syncthreads();

    const float scale = (float)NBINS / (ENT_HI - ENT_LO);
    float lsum[NCLS];
#pragma unroll
    for (int ci = 0; ci < NCLS; ++ci) lsum[ci] = 0.0f;

    const int stride = gridDim.x * blockDim.x;
    for (int j = blockIdx.x * blockDim.x + threadIdx.x; j < N; j += stride) {
        int t  = tgt[j];
        int bb = j / HW;
        int px = j - bb * HW;
        const float* p = prob + (size_t)bb * NCLS * HW + px;
        float ent = 0.0f;
#pragma unroll
        for (int ci = 0; ci < NCLS; ++ci) {
            float pv = p[(size_t)ci * HW];
            ent -= pv * __logf(pv + EPS);
        }
        if (STORE_ENT) ent_out[j] = ent;
        int bin = (int)((ent - ENT_LO) * scale);
        bin = min(max(bin, 0), NBINS - 1);
        atomicAdd(&lhist[t * NBINS + bin], 1);
#pragma unroll
        for (int ci = 0; ci < NCLS; ++ci)
            lsum[ci] += (t == ci) ? ent : 0.0f;
    }
    __syncthreads();
    for (int i = threadIdx.x; i < NCLS * NBINS; i += blockDim.x) {
        int v = lhist[i];
        if (v) atomicAdd(&ghist[i], v);
    }
#pragma unroll
    for (int ci = 0; ci < NCLS; ++ci)
        if (lsum[ci] != 0.0f) atomicAdd(&gsum[ci], lsum[ci]);
}

// Pass B: per-class percentile threshold from histogram (linear interpolation
// within bin approximates np.percentile's order-statistic lerp).
__global__ void k_thresh(const int* __restrict__ ghist, const float* __restrict__ gsum,
                         const int* __restrict__ dropp, float* __restrict__ thresh) {
    int cls = threadIdx.x;
    if (cls >= NCLS) return;
    const int* h = ghist + cls * NBINS;
    long long n = 0;
    for (int i = 0; i < NBINS; ++i) n += h[i];

    float th = INFINITY;  // no drops unless condition holds
    if (n > 0 && gsum[cls] > 10.0f) {
        float q   = (float)dropp[0] * 0.01f;
        float idx = q * (float)(n - 1);
        long long lo = (long long)floorf(idx);
        long long hi = (long long)ceilf(idx);
        float frac = idx - (float)lo;
        const float binw = (ENT_HI - ENT_LO) / (float)NBINS;
        float slo = ENT_HI, shi = ENT_HI;
        long long cum = 0;
        for (int i = 0; i < NBINS; ++i) {
            int cnt = h[i];
            if (cnt > 0) {
                long long cend = cum + cnt;
                if (lo >= cum && lo < cend)
                    slo = ENT_LO + ((float)i + ((float)(lo - cum) + 0.5f) / (float)cnt) * binw;
                if (hi >= cum && hi < cend)
                    shi = ENT_LO + ((float)i + ((float)(hi - cum) + 0.5f) / (float)cnt) * binw;
                cum = cend;
            }
        }
        th = slo * (1.0f - frac) + shi * frac;
    }
    thresh[cls] = th;
}

// Pass C: masked cross-entropy accumulation; WMMA-based wave reduction.
template <bool USE_ENT>
__global__ __launch_bounds__(256)
void k_loss(const float* __restrict__ pred, const int* __restrict__ tgt,
            const float* __restrict__ prob, const float* __restrict__ ent_in,
            const float* __restrict__ thresh, float* __restrict__ gacc, int N) {
    __shared__ float sth[NCLS];
    if (threadIdx.x < NCLS) sth[threadIdx.x] = thresh[threadIdx.x];
    __syncthreads();

    float lnll = 0.0f, lcnt = 0.0f;
    const int stride = gridDim.x * blockDim.x;
    for (int j = blockIdx.x * blockDim.x + threadIdx.x; j < N; j += stride) {
        int t  = tgt[j];
        int bb = j / HW;
        int px = j - bb * HW;

        float ent;
        if (USE_ENT) {
            ent = ent_in[j];
        } else {
            const float* p = prob + (size_t)bb * NCLS * HW + px;
            ent = 0.0f;
#pragma unroll
            for (int ci = 0; ci < NCLS; ++ci) {
                float pv = p[(size_t)ci * HW];
                ent -= pv * __logf(pv + EPS);
            }
        }
        float tth = 0.0f;
#pragma unroll
        for (int ci = 0; ci < NCLS; ++ci) tth = (t == ci) ? sth[ci] : tth;
        bool valid = ent < tth;

        const float* pp = pred + (size_t)bb * NCLS * HW + px;
        float pv[NCLS];
        float m = -INFINITY;
#pragma unroll
        for (int ci = 0; ci < NCLS; ++ci) {
            pv[ci] = pp[(size_t)ci * HW];
            m = fmaxf(m, pv[ci]);
        }
        float s = 0.0f, pt = 0.0f;
#pragma unroll
        for (int ci = 0; ci < NCLS; ++ci) {
            s += __expf(pv[ci] - m);
            pt = (t == ci) ? pv[ci] : pt;
        }
        float nll = (m + __logf(s)) - pt;
        if (valid) { lnll += nll; lcnt += 1.0f; }
    }

    float wnll = wave_sum_wmma(lnll);
    float wcnt = wave_sum_wmma(lcnt);
    if ((threadIdx.x & 31) == 0) {
        atomicAdd(&gacc[0], wnll);
        atomicAdd(&gacc[1], wcnt);
    }
}

__global__ void k_final(const float* __restrict__ gacc, float* __restrict__ out, int N) {
    if (threadIdx.x == 0 && blockIdx.x == 0) {
        float nv = gacc[1];
        out[0] = (nv > 0.0f) ? ((float)N * gacc[0] / (nv * nv)) : 0.0f;
    }
}

extern "C" void kernel_launch(void* const* d_in, const int* in_sizes, int n_in,
                              void* d_out, int out_size, void* d_ws, size_t ws_size,
                              hipStream_t stream) {
    const float* pred  = (const float*)d_in[0];
    const int*   tgt   = (const int*)d_in[1];
    const int*   dropp = (const int*)d_in[2];
    const float* prob  = (const float*)d_in[3];
    const int N = in_sizes[1];  // b*h*w

    // Workspace layout: [ent (optional, N floats)] [hist 5*2048 ints]
    //                   [gsum 5f] [thresh 5f] [gacc 2f]
    const size_t histBytes = (size_t)NCLS * NBINS * sizeof(int);
    const size_t metaBytes = histBytes + (2 * NCLS + 2) * sizeof(float);
    const size_t entBytes  = (size_t)N * sizeof(float);
    const bool useEnt = ws_size >= entBytes + metaBytes;

    char*  ws     = (char*)d_ws;
    float* ent    = (float*)ws;
    char*  meta   = useEnt ? (ws + entBytes) : ws;
    int*   hist   = (int*)meta;
    float* gsum   = (float*)(meta + histBytes);
    float* thresh = gsum + NCLS;
    float* gacc   = thresh + NCLS;

    const int zeroN = NCLS * NBINS + 2 * NCLS + 2;
    k_zero<<<(zeroN + 255) / 256, 256, 0, stream>>>((int*)meta, zeroN);

    const int blocks = 2048;  // 2048*256 divides N=2,097,152 -> uniform trip count
    if (useEnt)
        k_ent_hist<true><<<blocks, 256, 0, stream>>>(prob, tgt, ent, hist, gsum, N);
    else
        k_ent_hist<false><<<blocks, 256, 0, stream>>>(prob, tgt, nullptr, hist, gsum, N);

    k_thresh<<<1, 32, 0, stream>>>(hist, gsum, dropp, thresh);

    if (useEnt)
        k_loss<true><<<blocks, 256, 0, stream>>>(pred, tgt, prob, ent, thresh, gacc, N);
    else
        k_loss<false><<<blocks, 256, 0, stream>>>(pred, tgt, prob, nullptr, thresh, gacc, N);

    k_final<<<1, 32, 0, stream>>>(gacc, (float*)d_out, N);
}